// GAT_17849884082258
// MI455X (gfx1250) — compile-verified
//
#include <hip/hip_runtime.h>
#include <hip/hip_bf16.h>

#define D_H 128
#define NEG_SLOPE 0.2f

typedef __attribute__((ext_vector_type(2))) float v2f;
typedef __attribute__((ext_vector_type(8))) float v8f;

// ---------------------------------------------------------------------------
// Ordered-float helpers: monotonic float<->uint map so atomicMax(u32) works
// for signed floats (incl. -inf).
// ---------------------------------------------------------------------------
__device__ __forceinline__ unsigned f2ord(float f) {
  unsigned u = __float_as_uint(f);
  return (u & 0x80000000u) ? ~u : (u | 0x80000000u);
}
__device__ __forceinline__ float ord2f(unsigned u) {
  return __uint_as_float((u & 0x80000000u) ? (u & 0x7FFFFFFFu) : ~u);
}
#define ORD_NEG_INF 0x007FFFFFu  // f2ord(-inf)

__device__ __forceinline__ void edge_nodes(const int* __restrict__ ei, int i,
                                           int E, int& s, int& d) {
  if (i < E) { s = ei[i]; d = ei[E + i]; }
  else       { s = i - E; d = i - E; }      // self loops appended
}

// ---------------------------------------------------------------------------
// WMMA fp32 GEMM: out[M x 128] = A[M x 128] @ W[128 x 128] (+ bias)
// block = 256 threads = 8 waves. Block owns 16 rows; wave w owns cols
// [16w, 16w+16). K=128 -> 32 x v_wmma_f32_16x16x4_f32 per wave.
// Fragment layout per CDNA5 ISA 7.12.2 (32-bit A 16x4 / C 16x16 maps).
// ---------------------------------------------------------------------------
__global__ __launch_bounds__(256) void gemm128_wmma(
    const float* __restrict__ A, const float* __restrict__ W,
    const float* __restrict__ bias, float* __restrict__ out, int M) {
  __shared__ float lW[D_H * D_H];  // 64 KB: whole weight matrix in LDS
  {
    const float4* src = (const float4*)W;
    float4* dst = (float4*)lW;
    for (int i = threadIdx.x; i < (D_H * D_H) / 4; i += blockDim.x)
      dst[i] = src[i];
  }
  __syncthreads();

  const int lane  = threadIdx.x & 31;
  const int wave  = threadIdx.x >> 5;       // N-tile within block
  const int mbase = blockIdx.x * 16;
  const int nbase = wave * 16;
  const int halo  = lane >> 4;              // 0: lanes 0-15, 1: lanes 16-31
  const int l15   = lane & 15;

  int arow_i = mbase + l15;                 // clamp load row (stores guarded)
  if (arow_i >= M) arow_i = M - 1;
  const float* arow = A + (size_t)arow_i * D_H;
  const float* wcol = lW + nbase + l15;

  v8f c = {0.f, 0.f, 0.f, 0.f, 0.f, 0.f, 0.f, 0.f};

#pragma unroll
  for (int k = 0; k < D_H; k += 4) {
    const int kk = k + (halo << 1);         // lanes 0-15: K+0/K+1, 16-31: K+2/K+3
    const float2 av = *(const float2*)(arow + kk);   // contiguous -> b64 load
    v2f a; a.x = av.x; a.y = av.y;
    v2f b; b.x = wcol[kk * D_H]; b.y = wcol[(kk + 1) * D_H];
    c = __builtin_amdgcn_wmma_f32_16x16x4_f32(false, a, false, b,
                                              (short)0, c, false, false);
  }

  const int col = nbase + l15;
  const float bv = bias ? bias[col] : 0.f;
#pragma unroll
  for (int r = 0; r < 8; ++r) {             // VGPR r -> rows (halo*8 + r)
    const int row = mbase + halo * 8 + r;
    if (row < M) out[(size_t)row * D_H + col] = c[r] + bv;
  }
}

// ---------------------------------------------------------------------------
// Per-node attention scores: e_src[i] = ht[i,:]@a_src ; e_dst[i] = ht[i,:]@a_dst
// One wave32 per node, float4 per lane (32*4 = 128), shuffle tree reduce.
// ---------------------------------------------------------------------------
__global__ __launch_bounds__(256) void attn_dots_k(
    const float* __restrict__ ht, const float* __restrict__ a_src,
    const float* __restrict__ a_dst, float* __restrict__ e_src,
    float* __restrict__ e_dst, int n) {
  const int wid  = (blockIdx.x * blockDim.x + threadIdx.x) >> 5;
  const int lane = threadIdx.x & 31;
  if (wid >= n) return;
  const float4 hv = ((const float4*)(ht + (size_t)wid * D_H))[lane];
  const float4 as = ((const float4*)a_src)[lane];
  const float4 ad = ((const float4*)a_dst)[lane];
  float s = hv.x * as.x + hv.y * as.y + hv.z * as.z + hv.w * as.w;
  float d = hv.x * ad.x + hv.y * ad.y + hv.z * ad.z + hv.w * ad.w;
#pragma unroll
  for (int off = 16; off > 0; off >>= 1) {
    s += __shfl_down(s, off, 32);
    d += __shfl_down(d, off, 32);
  }
  if (lane == 0) { e_src[wid] = s; e_dst[wid] = d; }
}

// ---------------------------------------------------------------------------
// Per-layer init: segmax=-inf(ordered), denom=0, accum=0.
// ---------------------------------------------------------------------------
__global__ __launch_bounds__(256) void layer_init_k(
    unsigned* __restrict__ segmax, float* __restrict__ denom,
    float* __restrict__ accum, int n, int nfeat) {
  const int i = blockIdx.x * blockDim.x + threadIdx.x;
  if (i < n) { segmax[i] = ORD_NEG_INF; denom[i] = 0.f; }
  if (i < nfeat) accum[i] = 0.f;
}

// Pass 1: segment max of leaky_relu(e_src[s] + e_dst[d]) over dst.
__global__ __launch_bounds__(256) void edge_max_k(
    const int* __restrict__ ei, const float* __restrict__ es,
    const float* __restrict__ ed, unsigned* __restrict__ segmax,
    int E, int total) {
  const int i = blockIdx.x * blockDim.x + threadIdx.x;
  if (i >= total) return;
  int s, d; edge_nodes(ei, i, E, s, d);
  float e = es[s] + ed[d];
  e = (e > 0.f) ? e : NEG_SLOPE * e;
  atomicMax(segmax + d, f2ord(e));
}

// Pass 2: ex = exp(e - m[dst]); denom[dst] += ex.
__global__ __launch_bounds__(256) void edge_expsum_k(
    const int* __restrict__ ei, const float* __restrict__ es,
    const float* __restrict__ ed, const unsigned* __restrict__ segmax,
    float* __restrict__ denom, float* __restrict__ ex, int E, int total) {
  const int i = blockIdx.x * blockDim.x + threadIdx.x;
  if (i >= total) return;
  int s, d; edge_nodes(ei, i, E, s, d);
  float e = es[s] + ed[d];
  e = (e > 0.f) ? e : NEG_SLOPE * e;
  const float m = ord2f(segmax[d]);
  const float x = __expf(e - m);
  ex[i] = x;
  atomicAdd(denom + d, x);
}

// Pass 3: accum[dst,:] += (ex/denom[dst]) * ht[src,:]. One wave32 per edge,
// float4 gather per lane + 4 float atomics (all resident in L2).
__global__ __launch_bounds__(256) void edge_scatter_k(
    const int* __restrict__ ei, const float* __restrict__ ex,
    const float* __restrict__ denom, const float* __restrict__ ht,
    float* __restrict__ accum, int E, int total) {
  const int wid  = (blockIdx.x * blockDim.x + threadIdx.x) >> 5;
  const int lane = threadIdx.x & 31;
  if (wid >= total) return;
  int s, d; edge_nodes(ei, wid, E, s, d);
  const float alpha = ex[wid] / denom[d];
  const float4 hv = ((const float4*)(ht + (size_t)s * D_H))[lane];
  float* acc = accum + (size_t)d * D_H + lane * 4;
  atomicAdd(acc + 0, alpha * hv.x);
  atomicAdd(acc + 1, alpha * hv.y);
  atomicAdd(acc + 2, alpha * hv.z);
  atomicAdd(acc + 3, alpha * hv.w);
}

// h += accum + bias (residual + GAT bias), elementwise over N*128.
__global__ __launch_bounds__(256) void layer_finish_k(
    float* __restrict__ h, const float* __restrict__ accum,
    const float* __restrict__ bg, int nfeat) {
  const int i = blockIdx.x * blockDim.x + threadIdx.x;
  if (i >= nfeat) return;
  h[i] += accum[i] + bg[i & (D_H - 1)];
}

__global__ void zero_out_k(float* out, int n) {
  const int i = threadIdx.x;
  if (i < n) out[i] = 0.f;
}

// Final: out[j] += sigmoid(h[i,:] @ W2[:,j] + b2[j]); one wave per node.
__global__ __launch_bounds__(256) void final_k(
    const float* __restrict__ h, const float* __restrict__ W2,
    const float* __restrict__ b2, float* __restrict__ out, int n) {
  const int wid  = (blockIdx.x * blockDim.x + threadIdx.x) >> 5;
  const int lane = threadIdx.x & 31;
  if (wid >= n) return;
  const float4 hv = ((const float4*)(h + (size_t)wid * D_H))[lane];
  const int k0 = lane * 4;
  float s0 = hv.x * W2[(k0 + 0) * 2 + 0] + hv.y * W2[(k0 + 1) * 2 + 0] +
             hv.z * W2[(k0 + 2) * 2 + 0] + hv.w * W2[(k0 + 3) * 2 + 0];
  float s1 = hv.x * W2[(k0 + 0) * 2 + 1] + hv.y * W2[(k0 + 1) * 2 + 1] +
             hv.z * W2[(k0 + 2) * 2 + 1] + hv.w * W2[(k0 + 3) * 2 + 1];
#pragma unroll
  for (int off = 16; off > 0; off >>= 1) {
    s0 += __shfl_down(s0, off, 32);
    s1 += __shfl_down(s1, off, 32);
  }
  if (lane == 0) {
    s0 = 1.f / (1.f + __expf(-(s0 + b2[0])));
    s1 = 1.f / (1.f + __expf(-(s1 + b2[1])));
    atomicAdd(out + 0, s0);
    atomicAdd(out + 1, s1);
  }
}

// ---------------------------------------------------------------------------
extern "C" void kernel_launch(void* const* d_in, const int* in_sizes, int n_in,
                              void* d_out, int out_size, void* d_ws,
                              size_t ws_size, hipStream_t stream) {
  const float* x   = (const float*)d_in[0];
  const int*   ei  = (const int*)d_in[1];
  // d_in[2] = batch (unused)
  const float* W1  = (const float*)d_in[3];
  const float* b1  = (const float*)d_in[4];
  const float* Wg  = (const float*)d_in[5];
  const float* asr = (const float*)d_in[6];
  const float* ads = (const float*)d_in[7];
  const float* bg  = (const float*)d_in[8];
  const float* W2  = (const float*)d_in[9];
  const float* b2  = (const float*)d_in[10];

  const int N = in_sizes[0] / D_H;
  const int E = in_sizes[1] / 2;
  const int L = in_sizes[5] / (D_H * D_H);
  const int total = E + N;          // edges + self loops
  const int nfeat = N * D_H;

  // Workspace carve-up (floats): ~80 MB total.
  float* ws = (float*)d_ws;
  size_t o = 0;
  float*    h      = ws + o; o += (size_t)nfeat;
  float*    ht     = ws + o; o += (size_t)nfeat;
  float*    accum  = ws + o; o += (size_t)nfeat;
  float*    es     = ws + o; o += (size_t)N;
  float*    ed     = ws + o; o += (size_t)N;
  unsigned* segmax = (unsigned*)(ws + o); o += (size_t)N;
  float*    denom  = ws + o; o += (size_t)N;
  float*    exv    = ws + o; o += (size_t)total;

  const int mtiles      = (N + 15) / 16;
  const int blk_feat    = (nfeat + 255) / 256;
  const int blk_edges   = (total + 255) / 256;
  const int blk_nwaves  = (N * 32 + 255) / 256;       // one wave per node
  const int blk_ewaves  = ((size_t)total * 32 + 255) / 256;  // one wave per edge

  // h = x @ W1 + b1
  gemm128_wmma<<<mtiles, 256, 0, stream>>>(x, W1, b1, h, N);

  for (int l = 0; l < L; ++l) {
    layer_init_k<<<blk_feat, 256, 0, stream>>>(segmax, denom, accum, N, nfeat);
    // ht = h @ Wg[l]
    gemm128_wmma<<<mtiles, 256, 0, stream>>>(h, Wg + (size_t)l * D_H * D_H,
                                             nullptr, ht, N);
    attn_dots_k<<<blk_nwaves, 256, 0, stream>>>(ht, asr + l * D_H,
                                                ads + l * D_H, es, ed, N);
    edge_max_k<<<blk_edges, 256, 0, stream>>>(ei, es, ed, segmax, E, total);
    edge_expsum_k<<<blk_edges, 256, 0, stream>>>(ei, es, ed, segmax, denom,
                                                 exv, E, total);
    edge_scatter_k<<<blk_ewaves, 256, 0, stream>>>(ei, exv, denom, ht, accum,
                                                   E, total);
    layer_finish_k<<<blk_feat, 256, 0, stream>>>(h, accum, bg + l * D_H, nfeat);
  }

  zero_out_k<<<1, 32, 0, stream>>>((float*)d_out, out_size);
  final_k<<<blk_nwaves, 256, 0, stream>>>(h, W2, b2, (float*)d_out, N);
}